// RankLoss_2740189135200
// MI455X (gfx1250) — compile-verified
//
#include <hip/hip_runtime.h>
#include <math.h>

typedef __attribute__((ext_vector_type(2))) float v2f;
typedef __attribute__((ext_vector_type(8))) float v8f;

#define NPATCH 196
#define NPAD   208   // 13 tiles * 16
#define DIM    512
#define TEMPR  0.07f
#define EPSV   1e-8f

__device__ __forceinline__ float v8_get(v8f v, int i) {
    float r = v[0];
    if (i == 1) r = v[1];
    if (i == 2) r = v[2];
    if (i == 3) r = v[3];
    if (i == 4) r = v[4];
    if (i == 5) r = v[5];
    if (i == 6) r = v[6];
    if (i == 7) r = v[7];
    return r;
}

// One block (256 threads = 8 wave32) per batch sample.
__global__ void __launch_bounds__(256)
rankloss_per_sample(const float* __restrict__ patch,  // [B,196,512]
                    const float* __restrict__ text,   // [B,1,512]
                    const float* __restrict__ gt,     // [B,384,384]
                    float* __restrict__ per_sample)   // [B]
{
    const int b    = blockIdx.x;
    const int t    = threadIdx.x;
    const int lane = t & 31;
    const int wave = t >> 5;

    __shared__ float s_colmax[224];
    __shared__ float s_pos[NPATCH];
    __shared__ float s_dot[NPAD];
    __shared__ float s_n2[NPAD];
    __shared__ float s_sim[NPATCH];
    __shared__ float s_red[256];
    __shared__ float s_tn;

    // ---------------- Stage 1: pos flags from gathered max-pool ----------------
    const float* g = gt + (size_t)b * 384 * 384;
    for (int py = 0; py < 14; ++py) {
        if (t < 224) {
            const int sc = (t * 12) / 7;              // src[col]
            float cm = -3.0e38f;
            #pragma unroll
            for (int rr = 0; rr < 16; ++rr) {
                const int r  = py * 16 + rr;
                const int sr = (r * 12) / 7;          // src[row]
                cm = fmaxf(cm, g[sr * 384 + sc]);
            }
            s_colmax[t] = cm;
        }
        __syncthreads();
        if (t < 14) {
            float m = s_colmax[t * 16];
            #pragma unroll
            for (int k = 1; k < 16; ++k) m = fmaxf(m, s_colmax[t * 16 + k]);
            s_pos[py * 14 + t] = (m > 0.0f) ? 1.0f : 0.0f;
        }
        __syncthreads();
    }

    // ---------------- Stage 2: text norm ----------------
    const float* tx = text + (size_t)b * DIM;
    {
        float ts = 0.0f;
        for (int i = t; i < DIM; i += 256) { const float v = tx[i]; ts += v * v; }
        s_red[t] = ts;
        __syncthreads();
        for (int s = 128; s > 0; s >>= 1) {
            if (t < s) s_red[t] += s_red[t + s];
            __syncthreads();
        }
        if (t == 0) s_tn = fmaxf(sqrtf(s_red[0]), EPSV);
        __syncthreads();
    }

    // ---------------- Stage 3: WMMA dots (A x broadcast-text) + gram diag (A x A^T) ----
    // Padded rows (m >= 196) are clamped to row 195: their outputs land in
    // s_dot/s_n2[196..207] and off-diagonal gram entries, none of which are read.
    // No masking => EXEC stays all-1s (WMMA requirement) and loads are plain b64.
    const float* pe     = patch + (size_t)b * NPATCH * DIM;
    const int    m_lane = lane & 15;
    const int    khalf  = (lane >> 4) * 2;            // 0 or 2

    for (int tile = wave; tile < 13; tile += 8) {
        const int  m  = tile * 16 + m_lane;
        const int  mc = (m < NPATCH) ? m : (NPATCH - 1);
        const float* row = pe + (size_t)mc * DIM;

        v8f cd = {};   // dot accumulator (all 16 columns identical)
        v8f cg = {};   // gram accumulator (diagonal = squared norms)
        #pragma unroll 4
        for (int k = 0; k < DIM; k += 4) {
            const int kk = k + khalf;                 // even => 8B-aligned
            const v2f a  = *(const v2f*)(row + kk);   // A: K = kk, kk+1
            const v2f bb = *(const v2f*)(tx + kk);    // B: text broadcast across N
            cd = __builtin_amdgcn_wmma_f32_16x16x4_f32(false, a, false, bb,
                                                       (short)0, cd, false, false);
            cg = __builtin_amdgcn_wmma_f32_16x16x4_f32(false, a, false, a,
                                                       (short)0, cg, false, false);
        }
        // Extraction: lanes 0..7  -> m = lane     (VGPR lane,    N = lane, on-diagonal)
        //             lanes 24..31-> m = lane-16  (VGPR lane-24, N = lane-16)
        const int idx = lane & 7;
        const float dsel = v8_get(cd, idx);
        const float gsel = v8_get(cg, idx);
        if (lane < 8) {
            s_dot[tile * 16 + lane] = dsel;
            s_n2 [tile * 16 + lane] = gsel;
        } else if (lane >= 24) {
            s_dot[tile * 16 + lane - 16] = dsel;
            s_n2 [tile * 16 + lane - 16] = gsel;
        }
    }
    __syncthreads();

    // ---------------- Stage 4: similarities ----------------
    if (t < NPATCH) {
        const float pn  = fmaxf(sqrtf(s_n2[t]), EPSV);
        const float csv = s_dot[t] / (pn * s_tn);
        s_sim[t] = expf(csv / TEMPR);
    }
    __syncthreads();

    // ---------------- Stage 5: n_pos ----------------
    s_red[t] = (t < NPATCH) ? s_pos[t] : 0.0f;
    __syncthreads();
    for (int s = 128; s > 0; s >>= 1) {
        if (t < s) s_red[t] += s_red[t + s];
        __syncthreads();
    }
    const float n_pos = s_red[0];
    __syncthreads();

    // ---------------- Stage 6: pairwise hinge over pos_i & ~pos_j ----------------
    float acc = 0.0f;
    {
        const bool  is_j = (t < NPATCH);
        const float simj = is_j ? s_sim[t] : 0.0f;
        const bool  negj = is_j && (s_pos[t] == 0.0f);
        for (int i = 0; i < NPATCH; ++i) {
            if (s_pos[i] > 0.0f && negj) {
                acc += fmaxf(2.0f - s_sim[i] + simj, 0.0f);
            }
        }
    }
    s_red[t] = acc;
    __syncthreads();
    for (int s = 128; s > 0; s >>= 1) {
        if (t < s) s_red[t] += s_red[t + s];
        __syncthreads();
    }
    if (t == 0) {
        const float n_neg = 196.0f - n_pos;
        per_sample[b] = s_red[0] / (n_pos * n_neg);
    }
}

// Deterministic fixed-order mean over batch.
__global__ void __launch_bounds__(256)
rankloss_reduce(const float* __restrict__ per_sample, float* __restrict__ out, int nb)
{
    __shared__ float r[256];
    const int t = threadIdx.x;
    float a = 0.0f;
    for (int i = t; i < nb; i += 256) a += per_sample[i];
    r[t] = a;
    __syncthreads();
    for (int s = 128; s > 0; s >>= 1) {
        if (t < s) r[t] += r[t + s];
        __syncthreads();
    }
    if (t == 0) out[0] = r[0] / (float)nb;
}

extern "C" void kernel_launch(void* const* d_in, const int* in_sizes, int n_in,
                              void* d_out, int out_size, void* d_ws, size_t ws_size,
                              hipStream_t stream) {
    const float* patch = (const float*)d_in[0];   // [B,196,512] f32
    const float* text  = (const float*)d_in[1];   // [B,1,512]   f32
    const float* gt    = (const float*)d_in[2];   // [B,384,384] f32
    const int nb = in_sizes[1] / DIM;             // batch size (256)

    float* ps = (float*)d_ws;                     // nb floats of scratch

    rankloss_per_sample<<<nb, 256, 0, stream>>>(patch, text, gt, ps);
    rankloss_reduce<<<1, 256, 0, stream>>>(ps, (float*)d_out, nb);
}